// NodeModel_14216341750149
// MI455X (gfx1250) — compile-verified
//
#include <hip/hip_runtime.h>
#include <hip/hip_bf16.h>
#include <math.h>

typedef float v2f __attribute__((ext_vector_type(2)));
typedef float v4f __attribute__((ext_vector_type(4)));
typedef float v8f __attribute__((ext_vector_type(8)));

#define DD   64
#define HID  128
#define NEG_FLT_MAX (-3.402823466e38f)

// ---------------------------------------------------------------------------
// Kernel 1: init workspace (agg_sum = 0, agg_max = -FLT_MAX, cnt = 0)
// ---------------------------------------------------------------------------
__global__ __launch_bounds__(256) void init_kernel(float* __restrict__ agg_sum,
                                                   float* __restrict__ agg_max,
                                                   float* __restrict__ cnt,
                                                   int nd, int n) {
  int i = blockIdx.x * blockDim.x + threadIdx.x;
  if (i < nd) {
    agg_sum[i] = 0.0f;
    agg_max[i] = NEG_FLT_MAX;
  }
  if (i < n) cnt[i] = 0.0f;
}

// ---------------------------------------------------------------------------
// Kernel 2: edge scatter. 16 threads per edge, each handles 4 contiguous
// floats (non-temporal 16B load: edge_attr is streamed exactly once, keep it
// out of L2 so the 25.6 MB accumulator set stays resident). unsafeAtomic*
// guarantees native global_atomic_add_f32 / global_atomic_max_num_f32 (no
// CAS loops) resolving in L2.
// ---------------------------------------------------------------------------
__global__ __launch_bounds__(256) void scatter_kernel(const int* __restrict__ edge_index,
                                                      const float* __restrict__ edge_attr,
                                                      float* __restrict__ agg_sum,
                                                      float* __restrict__ agg_max,
                                                      float* __restrict__ cnt,
                                                      int n_edges) {
  long long t = (long long)blockIdx.x * blockDim.x + threadIdx.x;
  int e = (int)(t >> 4);
  int q = (int)(t & 15);
  if (e >= n_edges) return;
  int dst = edge_index[n_edges + e];  // row 1 of edge_index = destination (col)
  const v4f v =
      __builtin_nontemporal_load((const v4f*)(edge_attr + (size_t)e * DD + q * 4));
  float* sp = agg_sum + (size_t)dst * DD + q * 4;
  float* mp = agg_max + (size_t)dst * DD + q * 4;
  unsafeAtomicAdd(sp + 0, v.x);
  unsafeAtomicAdd(sp + 1, v.y);
  unsafeAtomicAdd(sp + 2, v.z);
  unsafeAtomicAdd(sp + 3, v.w);
  unsafeAtomicMax(mp + 0, v.x);
  unsafeAtomicMax(mp + 1, v.y);
  unsafeAtomicMax(mp + 2, v.z);
  unsafeAtomicMax(mp + 3, v.w);
  if (q == 0) unsafeAtomicAdd(cnt + dst, 1.0f);
}

// ---------------------------------------------------------------------------
// Kernel 3: fused feature-build + MLP (f32 WMMA) + LayerNorm + residual.
// One 128-thread block (4 waves) handles 16 nodes. W2 is DMA'd to LDS with
// global_load_async_to_lds_b128 at kernel entry and the wait is deferred to
// just before GEMM2 -> the copy is hidden under feature-build + GEMM1.
// ---------------------------------------------------------------------------
__device__ __forceinline__ float gelu_exact(float v) {
  return 0.5f * v * (1.0f + erff(v * 0.70710678118654752f));
}

__global__ __launch_bounds__(128) void mlp_kernel(const float* __restrict__ x,
                                                  const float* __restrict__ u,
                                                  const int* __restrict__ batch,
                                                  const float* __restrict__ W1,
                                                  const float* __restrict__ b1,
                                                  const float* __restrict__ W2,
                                                  const float* __restrict__ b2,
                                                  const float* __restrict__ gamma,
                                                  const float* __restrict__ beta,
                                                  const float* __restrict__ agg_sum,
                                                  const float* __restrict__ agg_max,
                                                  const float* __restrict__ cnt,
                                                  float* __restrict__ out, int n) {
  // Padded strides (260, 132, 68): stride % 64 == 4 -> fragment column reads
  // across 16 rows hit distinct LDS banks (conflict-free).
  __shared__ float Fs[16][260];      // feature tile, K = 0..255
  __shared__ float Hs[16][132];      // hidden activations (16 x 128)
  __shared__ float Os[16][68];       // pre-LN output (16 x 64)
  __shared__ float WS2[HID][DD + 4]; // W2 staged via async DMA (128 x 68)
  __shared__ float uval[16];
  __shared__ float mu_s[16], rs_s[16];

  const int node0 = blockIdx.x * 16;
  const int tid  = threadIdx.x;
  const int lane = tid & 31;
  const int wave = tid >> 5;          // 0..3

  // ---- Kick off async W2 -> LDS DMA (ASYNCcnt-tracked, waited before GEMM2)
  {
    unsigned long long w2base = (unsigned long long)(uintptr_t)W2;
    for (int i = tid; i < (HID * DD) / 4; i += 128) {  // 2048 x 16B chunks
      int r  = i >> 4;
      int c4 = (i & 15) * 4;
      // low 32 bits of a generic __shared__ address are the LDS offset
      unsigned ldsoff = (unsigned)(unsigned long long)(uintptr_t)(&WS2[r][c4]);
      unsigned goff   = (unsigned)((r * DD + c4) * sizeof(float));
      asm volatile("global_load_async_to_lds_b128 %0, %1, %2"
                   :: "v"(ldsoff), "v"(goff), "s"(w2base)
                   : "memory");
    }
  }

  // ---- Stage 1: build 16 x 256 feature tile in LDS --------------------------
  for (int i = tid; i < 16 * DD; i += 128) {
    int r = i >> 6, c = i & 63;
    int node = node0 + r;
    float xv = 0.f, s = 0.f, m = 0.f, cn = 0.f;
    if (node < n) {
      xv = x[(size_t)node * DD + c];
      s  = agg_sum[(size_t)node * DD + c];
      m  = agg_max[(size_t)node * DD + c];
      cn = cnt[node];
    }
    float mfix = (cn > 0.f) ? m : 0.f;      // empty segments -> 0
    float mean = s / fmaxf(cn, 1.f);
    Fs[r][c]        = xv;
    Fs[r][64 + c]   = s;
    Fs[r][128 + c]  = mfix;
    Fs[r][192 + c]  = mean;
  }
  if (tid < 16) {
    int node = node0 + tid;
    uval[tid] = (node < n) ? u[batch[node]] : 0.f;
  }
  __syncthreads();

  // Fragment layout constants (V_WMMA_F32_16X16X4_F32, wave32):
  //  A (16x4): lane L holds M = L%16, K = {kh, kh+1}, kh = 2*(L>>4)
  //  B (4x16): VGPR v holds K = v + kh, N = L%16
  //  C/D (16x16): VGPR r holds M = r + 8*(L>>4), N = L%16
  const int mrow  = lane & 15;
  const int khalf = (lane >> 4) * 2;
  const int ncol  = lane & 15;

  // ---- Stage 2: GEMM1 (16x256 @ 256x128), 2 column tiles per wave -----------
  v8f acc0 = {};
  v8f acc1 = {};
  const int c0 = (2 * wave) * 16 + ncol;   // hidden column for tile 0
  const int c1 = c0 + 16;                  // hidden column for tile 1
  for (int k = 0; k < 256; k += 4) {
    v2f a;
    a.x = Fs[mrow][k + khalf];
    a.y = Fs[mrow][k + khalf + 1];
    v2f bA, bB;
    bA.x = W1[(size_t)(k + khalf) * HID + c0];
    bA.y = W1[(size_t)(k + khalf + 1) * HID + c0];
    bB.x = W1[(size_t)(k + khalf) * HID + c1];
    bB.y = W1[(size_t)(k + khalf + 1) * HID + c1];
    acc0 = __builtin_amdgcn_wmma_f32_16x16x4_f32(false, a, false, bA, (short)0, acc0, false, false);
    acc1 = __builtin_amdgcn_wmma_f32_16x16x4_f32(false, a, false, bB, (short)0, acc1, false, false);
  }
  // Epilogue: + b1 + u * W1[256,:] (the K=257 rank-1 term), exact GELU
  {
    float w1u0 = W1[(size_t)256 * HID + c0];
    float w1u1 = W1[(size_t)256 * HID + c1];
    float bb0 = b1[c0], bb1 = b1[c1];
#pragma unroll
    for (int r = 0; r < 8; ++r) {
      int m = r + ((lane >> 4) * 8);
      float v0 = acc0[r] + bb0 + uval[m] * w1u0;
      float v1 = acc1[r] + bb1 + uval[m] * w1u1;
      Hs[m][c0] = gelu_exact(v0);
      Hs[m][c1] = gelu_exact(v1);
    }
  }
  // W2 DMA must be complete (and visible block-wide) before GEMM2 reads WS2.
  asm volatile("s_wait_asynccnt 0" ::: "memory");
  __syncthreads();

  // ---- Stage 3: GEMM2 (16x128 @ 128x64) from LDS, 1 column tile per wave ----
  v8f acc = {};
  const int oc = wave * 16 + ncol;
  for (int k = 0; k < HID; k += 4) {
    v2f a;
    a.x = Hs[mrow][k + khalf];
    a.y = Hs[mrow][k + khalf + 1];
    v2f b;
    b.x = WS2[k + khalf][oc];
    b.y = WS2[k + khalf + 1][oc];
    acc = __builtin_amdgcn_wmma_f32_16x16x4_f32(false, a, false, b, (short)0, acc, false, false);
  }
  {
    float bb = b2[oc];
#pragma unroll
    for (int r = 0; r < 8; ++r) {
      int m = r + ((lane >> 4) * 8);
      Os[m][oc] = acc[r] + bb;
    }
  }
  __syncthreads();

  // ---- Stage 4: LayerNorm over D=64 + gamma/beta + residual -----------------
  if (tid < 16) {
    float s = 0.f;
#pragma unroll
    for (int c = 0; c < DD; ++c) s += Os[tid][c];
    float mu = s * (1.0f / DD);
    float vs = 0.f;
#pragma unroll
    for (int c = 0; c < DD; ++c) {
      float d = Os[tid][c] - mu;
      vs += d * d;
    }
    mu_s[tid] = mu;
    rs_s[tid] = rsqrtf(vs * (1.0f / DD) + 1e-5f);
  }
  __syncthreads();
  for (int i = tid; i < 16 * DD; i += 128) {
    int r = i >> 6, c = i & 63;
    int node = node0 + r;
    if (node < n) {
      float v = (Os[r][c] - mu_s[r]) * rs_s[r] * gamma[c] + beta[c];
      out[(size_t)node * DD + c] = v + x[(size_t)node * DD + c];
    }
  }
}

// ---------------------------------------------------------------------------
extern "C" void kernel_launch(void* const* d_in, const int* in_sizes, int n_in,
                              void* d_out, int out_size, void* d_ws, size_t ws_size,
                              hipStream_t stream) {
  const float* x          = (const float*)d_in[0];
  const int*   edge_index = (const int*)d_in[1];
  const float* edge_attr  = (const float*)d_in[2];
  const float* u          = (const float*)d_in[3];
  const int*   batch      = (const int*)d_in[4];
  const float* W1         = (const float*)d_in[5];
  const float* b1         = (const float*)d_in[6];
  const float* W2         = (const float*)d_in[7];
  const float* b2         = (const float*)d_in[8];
  const float* gamma      = (const float*)d_in[9];
  const float* beta       = (const float*)d_in[10];
  float* out = (float*)d_out;

  const int n       = in_sizes[0] / DD;   // 50000
  const int n_edges = in_sizes[2] / DD;   // 1000000
  const int nd      = n * DD;

  float* ws      = (float*)d_ws;
  float* agg_sum = ws;                        // n*DD floats
  float* agg_max = ws + (size_t)nd;           // n*DD floats
  float* cnt     = ws + (size_t)2 * nd;       // n floats

  init_kernel<<<(nd + 255) / 256, 256, 0, stream>>>(agg_sum, agg_max, cnt, nd, n);

  long long sthreads = (long long)n_edges * 16;
  scatter_kernel<<<(unsigned)((sthreads + 255) / 256), 256, 0, stream>>>(
      edge_index, edge_attr, agg_sum, agg_max, cnt, n_edges);

  mlp_kernel<<<(n + 15) / 16, 128, 0, stream>>>(x, u, batch, W1, b1, W2, b2,
                                                gamma, beta, agg_sum, agg_max,
                                                cnt, out, n);
}